// ClassDistancePenaltyLoss_56788057588437
// MI455X (gfx1250) — compile-verified
//
#include <hip/hip_runtime.h>
#include <cstdint>

#define NUM_CLASSES 1000
#define FEAT        128
#define SLICES      8        // feature slices of 16
#define SLICE_F     16
#define ROW_CHUNKS  64
#define TILE        128      // rows per TDM tile
#define MAIN_THREADS 1024
#define ACC_STRIDE  17       // 17 coprime with 64 banks -> conflict-free LDS atomics
#define STAGE_STRIDE 17      // TDM pad: +1 dword per 16 dwords stored

typedef unsigned int v4u __attribute__((ext_vector_type(4)));
typedef int          v8i __attribute__((ext_vector_type(8)));
typedef int          v4i __attribute__((ext_vector_type(4)));
typedef float        v2f __attribute__((ext_vector_type(2)));
typedef float        v8f __attribute__((ext_vector_type(8)));

#if __has_builtin(__builtin_amdgcn_tensor_load_to_lds)
#define HAVE_TDM 1
#endif
#if __has_builtin(__builtin_amdgcn_wmma_f32_16x16x4_f32)
#define HAVE_WMMA_F32 1
#endif

// ---------------------------------------------------------------------------
// Kernel 1: zero workspace accumulators (gN[1000], gS[128000], gSS[128000])
// and the two output floats.
// ---------------------------------------------------------------------------
__global__ void zero_kernel(float* ws, float* out, int total) {
    int i = blockIdx.x * blockDim.x + threadIdx.x;
    if (i < total) ws[i] = 0.0f;
    if (i < 2) out[i] = 0.0f;
}

// ---------------------------------------------------------------------------
// Kernel 2: per-class counts (t is int64; read little-endian low dword).
// LDS-privatized integer counters, flushed with float global atomics.
// ---------------------------------------------------------------------------
__global__ void count_kernel(const unsigned* __restrict__ t2, float* __restrict__ gN, int n) {
    __shared__ unsigned cnt[NUM_CLASSES];
    for (int i = threadIdx.x; i < NUM_CLASSES; i += blockDim.x) cnt[i] = 0u;
    __syncthreads();
    int stride = gridDim.x * blockDim.x;
    for (int i = blockIdx.x * blockDim.x + threadIdx.x; i < n; i += stride) {
        unsigned c = t2[2 * (long long)i];           // low 32 bits of int64 label
        atomicAdd(&cnt[c], 1u);                      // ds_add_u32
    }
    __syncthreads();
    for (int i = threadIdx.x; i < NUM_CLASSES; i += blockDim.x) {
        unsigned v = cnt[i];
        if (v) atomicAdd(&gN[i], (float)v);          // global_atomic_add_f32
    }
}

// ---------------------------------------------------------------------------
// TDM helper: DMA a (rows x 16 f32) tile, row stride 128 f32, from global into
// LDS at lds_off, padding +1 dword per row (row stride 17 dwords in LDS).
// D# layout per CDNA5 ISA ch.8 (group0: 128b, group1: 256b; groups 2/3 zero
// for a 2D tensor).
// ---------------------------------------------------------------------------
#ifdef HAVE_TDM
__device__ __forceinline__ void tdm_load_tile(const float* gptr, unsigned lds_off, int rows) {
    unsigned long long ga = (unsigned long long)(uintptr_t)gptr;
    v4u g0;
    g0[0] = 1u;                                       // count=1 (valid user D#)
    g0[1] = lds_off;                                  // lds_addr (bytes)
    g0[2] = (unsigned)(ga & 0xFFFFFFFFu);             // global_addr[31:0]
    g0[3] = (unsigned)((ga >> 32) & 0x1FFFFFFu)       // global_addr[56:32]
          | (2u << 30);                               // type = 2 ("image")
    v8i g1;
    g1[0] = (2 << 16)                                 // data_size = 4 bytes
          | (1 << 20)                                 // pad_enable
          | (3 << 22);                                // pad_interval: 16 dwords
                                                      // pad_amount[31:25]=0 -> 1 dword
    g1[1] = (SLICE_F << 16);                          // tensor_dim0 = 16 (bits 79:48)
    g1[2] = (rows & 0xFFFF) << 16;                    // tensor_dim1 lo16 (bits 111:80)
    g1[3] = (SLICE_F << 16);                          // tile_dim0 = 16 (bits 127:112)
    g1[4] = rows & 0xFFFF;                            // tile_dim1 = rows; tile_dim2 = 0
    g1[5] = FEAT;                                     // tensor_dim0_stride = 128 elements
    g1[6] = 0;
    g1[7] = 0;
    v4i z4 = {0, 0, 0, 0};
    v8i z8 = {0, 0, 0, 0, 0, 0, 0, 0};
    __builtin_amdgcn_tensor_load_to_lds(g0, g1, z4, z4, z8, 0);
}
#endif

// ---------------------------------------------------------------------------
// Kernel 3 (main): streaming moment accumulation.
// grid = (ROW_CHUNKS, SLICES), block = 1024 threads (32 waves).
// Wave 0 drives a double-buffered TDM pipeline; all waves scatter-add into
// padded LDS accumulators with ds_add_f32; block flushes with global atomics.
// ---------------------------------------------------------------------------
__global__ __launch_bounds__(MAIN_THREADS, 1)
void moments_kernel(const float* __restrict__ x, const unsigned* __restrict__ t2,
                    float* __restrict__ gS, float* __restrict__ gSS, int n) {
    __shared__ float accS [NUM_CLASSES * ACC_STRIDE];
    __shared__ float accSS[NUM_CLASSES * ACC_STRIDE];
    __shared__ float stage[2][TILE * STAGE_STRIDE];

    const int tid   = threadIdx.x;
    const int slice = blockIdx.y;

    for (int i = tid; i < NUM_CLASSES * ACC_STRIDE; i += MAIN_THREADS) {
        accS[i]  = 0.0f;
        accSS[i] = 0.0f;
    }

    const int chunk = (n + ROW_CHUNKS - 1) / ROW_CHUNKS;
    const long long rowBeg = (long long)blockIdx.x * chunk;
    long long rem = (long long)n - rowBeg;
    if (rem <= 0) return;                              // uniform per block
    const int rows   = (rem < (long long)chunk) ? (int)rem : chunk;
    const int ntiles = (rows + TILE - 1) / TILE;

    const float* xbase = x + rowBeg * FEAT + slice * SLICE_F;
    const bool issuer  = (tid >> 5) == 0;              // wave 0 only issues TDM

#ifdef HAVE_TDM
    const unsigned lds0 = (unsigned)(uintptr_t)&stage[0][0];
    const unsigned lds1 = (unsigned)(uintptr_t)&stage[1][0];
    if (issuer) {
        int nr0 = rows < TILE ? rows : TILE;
        tdm_load_tile(xbase, lds0, nr0);
    }
#endif

    const int r  = tid & (TILE - 1);                   // row within tile (0..127)
    const int fg = tid >> 7;                           // feature pair group (0..7)

    for (int k = 0; k < ntiles; ++k) {
#ifdef HAVE_TDM
        if (issuer) {
            if (k + 1 < ntiles) {
                int left = rows - (k + 1) * TILE;
                int nr = left < TILE ? left : TILE;
                tdm_load_tile(xbase + (long long)(k + 1) * TILE * FEAT,
                              ((k + 1) & 1) ? lds1 : lds0, nr);
                __builtin_amdgcn_s_wait_tensorcnt(1);  // tile k landed
            } else {
                __builtin_amdgcn_s_wait_tensorcnt(0);
            }
        }
#endif
        __syncthreads();                               // release buffer k to all waves

        const int tr = (rows - k * TILE) < TILE ? (rows - k * TILE) : TILE;
        if (r < tr) {
            const long long row = rowBeg + (long long)k * TILE + r;
            const int cls = (int)t2[2 * row];          // low dword of int64 label
#ifdef HAVE_TDM
            const float* st = (k & 1) ? &stage[1][0] : &stage[0][0];
            float v0 = st[r * STAGE_STRIDE + fg * 2 + 0];
            float v1 = st[r * STAGE_STRIDE + fg * 2 + 1];
#else
            const float* gp = x + row * FEAT + slice * SLICE_F + fg * 2;
            float v0 = gp[0];
            float v1 = gp[1];
#endif
            const int a = cls * ACC_STRIDE + fg * 2;
            atomicAdd(&accS [a + 0], v0);              // ds_add_f32
            atomicAdd(&accS [a + 1], v1);
            atomicAdd(&accSS[a + 0], v0 * v0);
            atomicAdd(&accSS[a + 1], v1 * v1);
        }
        __syncthreads();                               // buffer k free for reload
    }

    // Flush LDS partials to global accumulators.
    for (int i = tid; i < NUM_CLASSES * SLICE_F; i += MAIN_THREADS) {
        int c = i >> 4, j = i & 15;
        int g = c * FEAT + slice * SLICE_F + j;
        atomicAdd(&gS [g], accS [c * ACC_STRIDE + j]);
        atomicAdd(&gSS[g], accSS[c * ACC_STRIDE + j]);
    }
}

// ---------------------------------------------------------------------------
// Kernel 4: variance + |.| + row-sum via V_WMMA_F32_16X16X4_F32 (B = ones so
// every D column equals the row sums; sum all lanes/VGPRs and divide by 16),
// then scalar mean over classes via one atomic per block.
// grid = ceil(1000/16) blocks x 32 threads (one wave).
// ---------------------------------------------------------------------------
__global__ void finalize_kernel(const float* __restrict__ gN, const float* __restrict__ gS,
                                const float* __restrict__ gSS, float* __restrict__ out) {
    const int lane = threadIdx.x;
    const int m    = lane & 15;                        // class row in tile
    const int half = lane >> 4;                        // K pair select (A-layout)
    const int c    = blockIdx.x * 16 + m;
    const bool valid = (c < NUM_CLASSES);

    float nv      = valid ? gN[c] : 2.0f;
    float inv_n   = 1.0f / nv;
    float inv_nm1 = 1.0f / (nv - 1.0f);

#ifdef HAVE_WMMA_F32
    v8f acc = {0.f, 0.f, 0.f, 0.f, 0.f, 0.f, 0.f, 0.f};
    v2f b;  b.x = 1.0f;  b.y = 1.0f;
    for (int kc = 0; kc < 32; ++kc) {
        int f0 = kc * 4 + half * 2;                    // A 16x4 f32 lane layout
        v2f a;
        if (valid) {
            float s0  = gS [c * FEAT + f0 + 0], q0 = gSS[c * FEAT + f0 + 0];
            float s1  = gS [c * FEAT + f0 + 1], q1 = gSS[c * FEAT + f0 + 1];
            a.x = fabsf((q0 - s0 * s0 * inv_n) * inv_nm1);
            a.y = fabsf((q1 - s1 * s1 * inv_n) * inv_nm1);
        } else {
            a.x = 0.0f; a.y = 0.0f;
        }
        acc = __builtin_amdgcn_wmma_f32_16x16x4_f32(false, a, false, b,
                                                    (short)0, acc, false, false);
    }
    float partial = 0.0f;
    for (int i = 0; i < 8; ++i) partial += acc[i];
    for (int off = 16; off > 0; off >>= 1) partial += __shfl_down(partial, off);
    if (lane == 0)
        atomicAdd(out, partial * (1.0f / 16.0f) * (1.0f / (float)NUM_CLASSES));
#else
    float partial = 0.0f;
    for (int kc = 0; kc < 32; ++kc) {
        int f0 = kc * 4 + half * 2;
        if (valid) {
            float s0 = gS [c * FEAT + f0 + 0], q0 = gSS[c * FEAT + f0 + 0];
            float s1 = gS [c * FEAT + f0 + 1], q1 = gSS[c * FEAT + f0 + 1];
            partial += fabsf((q0 - s0 * s0 * inv_n) * inv_nm1);
            partial += fabsf((q1 - s1 * s1 * inv_n) * inv_nm1);
        }
    }
    for (int off = 16; off > 0; off >>= 1) partial += __shfl_down(partial, off);
    if (lane == 0) atomicAdd(out, partial * (1.0f / (float)NUM_CLASSES));
#endif
}

// ---------------------------------------------------------------------------
extern "C" void kernel_launch(void* const* d_in, const int* in_sizes, int n_in,
                              void* d_out, int out_size, void* d_ws, size_t ws_size,
                              hipStream_t stream) {
    const float*    x  = (const float*)d_in[0];
    const unsigned* t2 = (const unsigned*)d_in[1];     // int64 labels, read low dwords
    float* out = (float*)d_out;

    float* gN  = (float*)d_ws;                         // [1000]
    float* gS  = gN + NUM_CLASSES;                     // [1000*128]
    float* gSS = gS + NUM_CLASSES * FEAT;              // [1000*128]
    const int acc_total = NUM_CLASSES + 2 * NUM_CLASSES * FEAT;   // 257,000 f32 (~1 MB)

    const int n = in_sizes[0] / FEAT;                  // 2,000,000 rows

    zero_kernel<<<(acc_total + 255) / 256, 256, 0, stream>>>(gN, out, acc_total);
    count_kernel<<<512, 256, 0, stream>>>(t2, gN, n);

    dim3 grid(ROW_CHUNKS, SLICES);
    moments_kernel<<<grid, MAIN_THREADS, 0, stream>>>(x, t2, gS, gSS, n);

    finalize_kernel<<<(NUM_CLASSES + 15) / 16, 32, 0, stream>>>(gN, gS, gSS, out);
}